// Replicator2Layer_10831907520900
// MI455X (gfx1250) — compile-verified
//
#include <hip/hip_runtime.h>

// Problem constants (match reference: B=4, N=256, E=512, fp32 throughout)
#define BATCH 4
#define NPOS  256
#define EMB   512

typedef float v2f __attribute__((ext_vector_type(2)));
typedef float v8f __attribute__((ext_vector_type(8)));

// D = A(16x4 f32) * B(4x16 f32) + C(16x16 f32)   -- CDNA5 V_WMMA_F32_16X16X4_F32
static __device__ __forceinline__ v8f wmma4_f32(v2f a, v2f b, v8f c) {
  return __builtin_amdgcn_wmma_f32_16x16x4_f32(
      /*neg_a=*/false, a, /*neg_b=*/false, b,
      /*c_mod=*/(short)0, c, /*reuse_a=*/false, /*reuse_b=*/false);
}

// Store 16x16 f32 C/D fragment (lane<16 -> N=lane, M=j; lane>=16 -> N=lane-16,
// M=j+8) to row-major out with leading dimension ld.
static __device__ __forceinline__ void store_cd(v8f acc, float* __restrict__ out,
                                                int i0, int n0, int ld, int lane) {
  const int ncol  = lane & 15;
  const int mbase = (lane >> 4) << 3;
#pragma unroll
  for (int j = 0; j < 8; ++j)
    out[(size_t)(i0 + mbase + j) * ld + (n0 + ncol)] = acc[j];
}

// ---------------------------------------------------------------------------
// G[b,i,n] = sum_e X[b,i,e] * X[b,n,e]      (X·X^T, K=512)
// 1 wave per 16x16 tile; 4*16*16 = 1024 waves
// ---------------------------------------------------------------------------
__global__ __launch_bounds__(256) void k_gram(const float* __restrict__ X,
                                              float* __restrict__ G) {
  const int wave = blockIdx.x * 8 + (threadIdx.x >> 5);
  const int lane = threadIdx.x & 31;
  const int b  = wave >> 8;
  const int t  = wave & 255;
  const int i0 = (t >> 4) << 4;
  const int n0 = (t & 15) << 4;

  const float* Xb = X + (size_t)b * NPOS * EMB;
  const int r  = lane & 15;          // A-row / B-col within tile
  const int kh = (lane >> 4) << 1;   // K sub-offset: 0 (lanes 0-15) or 2 (16-31)
  const float* arow = Xb + (size_t)(i0 + r) * EMB + kh;
  const float* brow = Xb + (size_t)(n0 + r) * EMB + kh;   // X^T column = X row

  v8f acc = {0.f, 0.f, 0.f, 0.f, 0.f, 0.f, 0.f, 0.f};
#pragma unroll 8
  for (int k = 0; k < EMB; k += 4) {
    v2f a  = *(const v2f*)(arow + k);
    v2f bb = *(const v2f*)(brow + k);
    acc = wmma4_f32(a, bb, acc);
  }
  store_cd(acc, G + (size_t)b * NPOS * NPOS, i0, n0, NPOS, lane);
}

// ---------------------------------------------------------------------------
// S[b,i,p] = sum_{n<=i} G[b,i,n] * Kw[n,p]   (tril(G)·Kw)
// Main loop: full 16-wide K blocks below the diagonal block (mask-free,
// fully unrolled: 4 WMMA + 8 loads). Tail: one masked diagonal block.
// ---------------------------------------------------------------------------
__global__ __launch_bounds__(256) void k_skw(const float* __restrict__ G,
                                             const float* __restrict__ Kw,
                                             float* __restrict__ S) {
  const int wave = blockIdx.x * 8 + (threadIdx.x >> 5);
  const int lane = threadIdx.x & 31;
  const int b  = wave >> 8;
  const int t  = wave & 255;
  const int i0 = (t >> 4) << 4;
  const int p0 = (t & 15) << 4;

  const int r  = lane & 15;
  const int kh = (lane >> 4) << 1;
  const int i  = i0 + r;
  const float* arow = G + (size_t)b * NPOS * NPOS + (size_t)i * NPOS;
  const float* bcol = Kw + p0 + r;

  v8f acc = {0.f, 0.f, 0.f, 0.f, 0.f, 0.f, 0.f, 0.f};
  for (int kb = 0; kb < i0; kb += 16) {        // strictly-below-diagonal blocks
#pragma unroll
    for (int u = 0; u < 4; ++u) {
      const int k0 = kb + (u << 2) + kh;
      v2f a  = *(const v2f*)(arow + k0);
      v2f bb = { bcol[(size_t)k0 * NPOS], bcol[(size_t)(k0 + 1) * NPOS] };
      acc = wmma4_f32(a, bb, acc);
    }
  }
#pragma unroll
  for (int u = 0; u < 4; ++u) {                // diagonal block: tril mask
    const int k0 = i0 + (u << 2) + kh;
    float a0 = arow[k0];
    float a1 = arow[k0 + 1];
    a0 = (k0     <= i) ? a0 : 0.f;
    a1 = (k0 + 1 <= i) ? a1 : 0.f;
    v2f a  = {a0, a1};
    v2f bb = { bcol[(size_t)k0 * NPOS], bcol[(size_t)(k0 + 1) * NPOS] };
    acc = wmma4_f32(a, bb, acc);
  }
  store_cd(acc, S + (size_t)b * NPOS * NPOS, i0, p0, NPOS, lane);
}

// ---------------------------------------------------------------------------
// T[b,i,n] = sum_{p<=i} S[b,i,p] * Qw[n,p]   (tril(S)·Qw^T)
// B fragment is a contiguous float2 from Qw row n  (B[p][n] = Qw[n][p])
// ---------------------------------------------------------------------------
__global__ __launch_bounds__(256) void k_tqw(const float* __restrict__ S,
                                             const float* __restrict__ Qw,
                                             float* __restrict__ T) {
  const int wave = blockIdx.x * 8 + (threadIdx.x >> 5);
  const int lane = threadIdx.x & 31;
  const int b  = wave >> 8;
  const int t  = wave & 255;
  const int i0 = (t >> 4) << 4;
  const int n0 = (t & 15) << 4;

  const int r  = lane & 15;
  const int kh = (lane >> 4) << 1;
  const int i  = i0 + r;
  const float* arow = S + (size_t)b * NPOS * NPOS + (size_t)i * NPOS;
  const float* qrow = Qw + (size_t)(n0 + r) * NPOS + kh;

  v8f acc = {0.f, 0.f, 0.f, 0.f, 0.f, 0.f, 0.f, 0.f};
  for (int kb = 0; kb < i0; kb += 16) {
#pragma unroll
    for (int u = 0; u < 4; ++u) {
      const int ku = kb + (u << 2);
      v2f a  = *(const v2f*)(arow + ku + kh);
      v2f bb = *(const v2f*)(qrow + ku);
      acc = wmma4_f32(a, bb, acc);
    }
  }
#pragma unroll
  for (int u = 0; u < 4; ++u) {
    const int ku = i0 + (u << 2);
    const int k0 = ku + kh;
    float a0 = arow[k0];
    float a1 = arow[k0 + 1];
    a0 = (k0     <= i) ? a0 : 0.f;
    a1 = (k0 + 1 <= i) ? a1 : 0.f;
    v2f a  = {a0, a1};
    v2f bb = *(const v2f*)(qrow + ku);
    acc = wmma4_f32(a, bb, acc);
  }
  store_cd(acc, T + (size_t)b * NPOS * NPOS, i0, n0, NPOS, lane);
}

// ---------------------------------------------------------------------------
// F[b,i,m] = sum_{n<=i} T[b,i,n] * X[b,n,m]   (tril(T)·X, 256x512 out)
// 4 * 16 * 32 = 2048 waves
// ---------------------------------------------------------------------------
__global__ __launch_bounds__(256) void k_fit(const float* __restrict__ T,
                                             const float* __restrict__ X,
                                             float* __restrict__ F) {
  const int wave = blockIdx.x * 8 + (threadIdx.x >> 5);
  const int lane = threadIdx.x & 31;
  const int b  = wave >> 9;
  const int t  = wave & 511;
  const int i0 = (t >> 5) << 4;
  const int m0 = (t & 31) << 4;

  const int r  = lane & 15;
  const int kh = (lane >> 4) << 1;
  const int i  = i0 + r;
  const float* arow = T + (size_t)b * NPOS * NPOS + (size_t)i * NPOS;
  const float* bcol = X + (size_t)b * NPOS * EMB + m0 + r;

  v8f acc = {0.f, 0.f, 0.f, 0.f, 0.f, 0.f, 0.f, 0.f};
  for (int kb = 0; kb < i0; kb += 16) {
#pragma unroll
    for (int u = 0; u < 4; ++u) {
      const int k0 = kb + (u << 2) + kh;
      v2f a  = *(const v2f*)(arow + k0);
      v2f bb = { bcol[(size_t)k0 * EMB], bcol[(size_t)(k0 + 1) * EMB] };
      acc = wmma4_f32(a, bb, acc);
    }
  }
#pragma unroll
  for (int u = 0; u < 4; ++u) {
    const int k0 = i0 + (u << 2) + kh;
    float a0 = arow[k0];
    float a1 = arow[k0 + 1];
    a0 = (k0     <= i) ? a0 : 0.f;
    a1 = (k0 + 1 <= i) ? a1 : 0.f;
    v2f a  = {a0, a1};
    v2f bb = { bcol[(size_t)k0 * EMB], bcol[(size_t)(k0 + 1) * EMB] };
    acc = wmma4_f32(a, bb, acc);
  }
  store_cd(acc, F + (size_t)b * NPOS * EMB, i0, m0, EMB, lane);
}

// ---------------------------------------------------------------------------
// avg[b,i] = <x_i, fit_i>;  out = x * (1 + fit - avg)
// one 256-thread block per (b,i); each thread handles 2 of 512 elements
// ---------------------------------------------------------------------------
__global__ __launch_bounds__(256) void k_final(const float* __restrict__ X,
                                               const float* __restrict__ F,
                                               float* __restrict__ Out) {
  __shared__ float red[256];
  const int b = blockIdx.x >> 8;
  const int i = blockIdx.x & 255;
  const size_t base = ((size_t)b * NPOS + i) * EMB;
  const int t = threadIdx.x;

  const float x0 = X[base + t];
  const float x1 = X[base + t + 256];
  const float f0 = F[base + t];
  const float f1 = F[base + t + 256];

  red[t] = x0 * f0 + x1 * f1;
  __syncthreads();
#pragma unroll
  for (int s = 128; s > 0; s >>= 1) {
    if (t < s) red[t] += red[t + s];
    __syncthreads();
  }
  const float avg = red[0];
  Out[base + t]       = x0 * (1.0f + f0 - avg);
  Out[base + t + 256] = x1 * (1.0f + f1 - avg);
}

extern "C" void kernel_launch(void* const* d_in, const int* in_sizes, int n_in,
                              void* d_out, int out_size, void* d_ws, size_t ws_size,
                              hipStream_t stream) {
  const float* X  = (const float*)d_in[0];   // (4,256,512)
  const float* Qw = (const float*)d_in[1];   // (256,256)
  const float* Kw = (const float*)d_in[2];   // (256,256)
  float* out = (float*)d_out;                // (4,256,512)

  // Workspace layout (fp32): G | S | T | F  = 1MB + 1MB + 1MB + 2MB = 5MB
  float* G = (float*)d_ws;
  float* S = G + (size_t)BATCH * NPOS * NPOS;
  float* T = S + (size_t)BATCH * NPOS * NPOS;
  float* F = T + (size_t)BATCH * NPOS * NPOS;

  // 1024 tile-waves -> 128 blocks of 8 waves (256 threads, wave32)
  k_gram <<<128,  256, 0, stream>>>(X, G);
  k_skw  <<<128,  256, 0, stream>>>(G, Kw, S);
  k_tqw  <<<128,  256, 0, stream>>>(S, Qw, T);
  k_fit  <<<256,  256, 0, stream>>>(T, X, F);   // 2048 waves
  k_final<<<1024, 256, 0, stream>>>(X, F, out);
}